// MPNN_26465588478228
// MI455X (gfx1250) — compile-verified
//
#include <hip/hip_runtime.h>

typedef __attribute__((ext_vector_type(2))) float v2f;
typedef __attribute__((ext_vector_type(8))) float v8f;
typedef __attribute__((ext_vector_type(4))) unsigned int v4u;
typedef __attribute__((ext_vector_type(8))) int v8i;
typedef __attribute__((ext_vector_type(4))) int v4i;

#define Nn 50000
#define Ee 600000
#define Dd 128
#define Hh 128
#define Cc 40
#define Ll 3
#define EPS 1e-5f

// ---------------- degree / dinv ----------------
__global__ __launch_bounds__(256) void k_deg_init(float* deg) {
    int i = blockIdx.x * 256 + threadIdx.x;
    if (i < Nn) deg[i] = 1.0f;  // self loop
}

__global__ __launch_bounds__(256) void k_deg_count(const int* __restrict__ dst, float* deg) {
    int e = blockIdx.x * 256 + threadIdx.x;
    if (e < Ee) unsafeAtomicAdd(&deg[dst[e]], 1.0f);
}

__global__ __launch_bounds__(256) void k_dinv(const float* __restrict__ deg, float* dinv) {
    int i = blockIdx.x * 256 + threadIdx.x;
    if (i < Nn) dinv[i] = rsqrtf(deg[i]);
}

// ---------------- weight pre-swizzle into per-lane WMMA B layout ----------------
// Wb[mat][ntile][kt][lane][2]: one contiguous 256B block per wave per K-step.
__global__ __launch_bounds__(256) void k_wswizzle(
    const float* __restrict__ Wc, const float* __restrict__ Wl, float* __restrict__ Wb)
{
    int gid = blockIdx.x * 256 + threadIdx.x;   // 6 * 8192
    if (gid >= 6 * 8192) return;
    int mat  = gid >> 13;          // 0..5  (layer*2 + which)
    int r    = gid & 8191;
    int nt   = r >> 10;            // 0..7
    int kt   = (r >> 5) & 31;      // 0..31
    int lane = r & 31;
    int l = mat >> 1;
    const float* W = ((mat & 1) ? Wl : Wc) + (size_t)l * Dd * Hh;
    int k0  = kt * 4 + 2 * (lane >> 4);
    int col = nt * 16 + (lane & 15);
    float2 v;
    v.x = W[(size_t)k0 * Hh + col];
    v.y = W[(size_t)(k0 + 1) * Hh + col];
    *(float2*)(Wb + (size_t)mat * 16384 +
               ((size_t)(nt * 32 + kt) * 32 + lane) * 2) = v;
}

// ---------------- fused dual GEMM + self-loop epilogue ----------------
// Hc = X*Wc ; Hl = X*Wl + bl ; Agg = Hc*dinv^2 + bconv
// block = 256 threads (8 waves). Each block: 16 rows of X. Each wave: 16 cols.
__global__ __launch_bounds__(256) void k_gemm_dual(
    const float* __restrict__ X,
    const float* __restrict__ WbC, const float* __restrict__ WbL,
    const float* __restrict__ bl, const float* __restrict__ bconv,
    const float* __restrict__ dinv,
    float* __restrict__ Hc, float* __restrict__ Hl, float* __restrict__ Agg)
{
    __shared__ float lds[16 * 132];  // A tile, padded stride -> conflict-free
    const int row0 = blockIdx.x * 16;
    const int t = threadIdx.x;

#if __has_builtin(__builtin_amdgcn_tensor_load_to_lds) && \
    __has_builtin(__builtin_amdgcn_s_wait_tensorcnt)
    // Tensor Data Mover: DMA the 16x128 fp32 tile into LDS, hardware-padding
    // each 128-DWORD row by 4 DWORDs (-> 132-float LDS stride).
    if (t < 32) {  // wave 0 issues the descriptor
        unsigned long long gaddr =
            (unsigned long long)(size_t)(X + (size_t)row0 * Dd);
        unsigned int lds_off = (unsigned int)(size_t)(&lds[0]);
        v4u g0;
        g0[0] = 1u;                                        // count=1, user D#
        g0[1] = lds_off;                                   // lds_addr
        g0[2] = (unsigned int)(gaddr & 0xFFFFFFFFu);       // global_addr lo
        g0[3] = (unsigned int)((gaddr >> 32) & 0x1FFFFFFu) // global_addr hi
                | (2u << 30);                              // type = 2 (image)
        v8i g1;
        g1[0] = (2 << 16)    // data_size = 4B
              | (1 << 20)    // pad_enable
              | (6 << 22)    // pad_interval: 128 DWORDs
              | (3 << 25);   // pad_amount:  4 DWORDs
        g1[1] = (int)(128u << 16);   // tensor_dim0 = 128
        g1[2] = (int)(16u << 16);    // tensor_dim1 = 16
        g1[3] = (int)(128u << 16);   // tile_dim0 = 128
        g1[4] = 16;                  // tile_dim1 = 16
        g1[5] = 128;                 // tensor_dim0_stride = 128
        g1[6] = 0;
        g1[7] = 0;
        v4i z4 = {0, 0, 0, 0};
        v8i z8 = {0, 0, 0, 0, 0, 0, 0, 0};
        __builtin_amdgcn_tensor_load_to_lds(g0, g1, z4, z4, z8, 0);
        __builtin_amdgcn_s_wait_tensorcnt(0);
    }
    __syncthreads();
#else
    {   // fallback: cooperative vector load of the A tile
        int r = t >> 4;
        int c = (t & 15) * 8;
        const float* sp = X + (size_t)(row0 + r) * Dd + c;
        float4 a0 = *(const float4*)(sp);
        float4 a1 = *(const float4*)(sp + 4);
        float* dp = &lds[r * 132 + c];
        *(float4*)(dp) = a0;
        *(float4*)(dp + 4) = a1;
    }
    __syncthreads();
#endif

    const int wave = t >> 5;          // 0..7 -> N tile
    const int lane = t & 31;
    const int n0 = wave * 16;
    const int m = lane & 15;          // A row within tile
    const int khalf = (lane >> 4) * 2;
    const int ncol = lane & 15;       // B/C column within tile

    const float* wbc = WbC + (size_t)wave * 2048 + lane * 2;
    const float* wbl = WbL + (size_t)wave * 2048 + lane * 2;

    v8f accC = {};
    v8f accL = {};
#pragma unroll 4
    for (int kt = 0; kt < 32; ++kt) {
        const int k0 = kt * 4 + khalf;
        v2f a;
        a.x = lds[m * 132 + k0];
        a.y = lds[m * 132 + k0 + 1];
        v2f bc = *(const v2f*)(wbc + kt * 64);   // contiguous 256B per wave
        v2f bL = *(const v2f*)(wbl + kt * 64);
        accC = __builtin_amdgcn_wmma_f32_16x16x4_f32(false, a, false, bc,
                                                     (short)0, accC, false, false);
        accL = __builtin_amdgcn_wmma_f32_16x16x4_f32(false, a, false, bL,
                                                     (short)0, accL, false, false);
    }

    const float bias = bl[n0 + ncol];
    const float bcv  = bconv[n0 + ncol];
    const int rbase = row0 + 8 * (lane >> 4);
#pragma unroll
    for (int j = 0; j < 8; ++j) {
        const int r = rbase + j;
        const float di = dinv[r];
        const size_t idx = (size_t)r * Hh + n0 + ncol;
        const float hc = accC[j];
        Hc[idx]  = hc;
        Agg[idx] = fmaf(hc, di * di, bcv);       // self-loop + bconv
        Hl[idx]  = accL[j] + bias;
    }
}

// ---------------- edge scatter: agg[dst] += h[src] * dinv[src]*dinv[dst] ----------------
// one wave per edge, float4 per lane (contiguous 512B gather per wave)
__global__ __launch_bounds__(256) void k_edge_scatter(
    const int* __restrict__ src, const int* __restrict__ dst,
    const float* __restrict__ dinv, const float* __restrict__ Hc,
    float* __restrict__ agg)
{
    int gid = blockIdx.x * 256 + threadIdx.x;  // over E*32
    int e = gid >> 5;
    int c = (gid & 31) * 4;
    if (e >= Ee) return;
    int s = src[e];
    int d = dst[e];
    float w = dinv[s] * dinv[d];
    float4 h = *(const float4*)(Hc + (size_t)s * Hh + c);
    float* p = agg + (size_t)d * Hh + c;
    unsafeAtomicAdd(p + 0, h.x * w);
    unsafeAtomicAdd(p + 1, h.y * w);
    unsafeAtomicAdd(p + 2, h.z * w);
    unsafeAtomicAdd(p + 3, h.w * w);
}

// ---------------- LayerNorm + ReLU + JK sum: one wave32 per node ----------------
__global__ __launch_bounds__(256) void k_post_ln(
    const float* __restrict__ agg, const float* __restrict__ Hl,
    const float* __restrict__ g, const float* __restrict__ b,
    float* __restrict__ xcur, float* __restrict__ xfin, int first)
{
    int wave = threadIdx.x >> 5;
    int lane = threadIdx.x & 31;
    int node = blockIdx.x * 8 + wave;
    if (node >= Nn) return;
    size_t base = (size_t)node * Hh + lane * 4;
    float4 v = *(const float4*)(agg + base);
    float4 l = *(const float4*)(Hl + base);
    v.x += l.x; v.y += l.y; v.z += l.z; v.w += l.w;

    float s = v.x + v.y + v.z + v.w;
    float ss = v.x * v.x + v.y * v.y + v.z * v.z + v.w * v.w;
#pragma unroll
    for (int off = 16; off > 0; off >>= 1) {
        s  += __shfl_xor(s, off, 32);
        ss += __shfl_xor(ss, off, 32);
    }
    float mean = s * (1.0f / Hh);
    float var = ss * (1.0f / Hh) - mean * mean;
    float inv = rsqrtf(var + EPS);

    float4 gv = *(const float4*)(g + lane * 4);
    float4 bv = *(const float4*)(b + lane * 4);
    v.x = fmaxf(0.0f, (v.x - mean) * inv * gv.x + bv.x);
    v.y = fmaxf(0.0f, (v.y - mean) * inv * gv.y + bv.y);
    v.z = fmaxf(0.0f, (v.z - mean) * inv * gv.z + bv.z);
    v.w = fmaxf(0.0f, (v.w - mean) * inv * gv.w + bv.w);

    *(float4*)(xcur + base) = v;
    if (first) {
        *(float4*)(xfin + base) = v;
    } else {
        float4 f = *(const float4*)(xfin + base);
        f.x += v.x; f.y += v.y; f.z += v.z; f.w += v.w;
        *(float4*)(xfin + base) = f;
    }
}

// ---------------- prediction head: out = xfin @ Wpred + bpred ----------------
__global__ __launch_bounds__(64) void k_pred(
    const float* __restrict__ xf, const float* __restrict__ Wp,
    const float* __restrict__ bp, float* __restrict__ out)
{
    __shared__ float row[Hh];
    int n = blockIdx.x;
    int t = threadIdx.x;
    float2 v = *(const float2*)(xf + (size_t)n * Hh + t * 2);
    row[t * 2] = v.x;
    row[t * 2 + 1] = v.y;
    __syncthreads();
    if (t < Cc) {
        float acc = bp[t];
#pragma unroll 8
        for (int k = 0; k < Hh; ++k)
            acc = fmaf(row[k], Wp[k * Cc + t], acc);
        out[(size_t)n * Cc + t] = acc;
    }
}

extern "C" void kernel_launch(void* const* d_in, const int* in_sizes, int n_in,
                              void* d_out, int out_size, void* d_ws, size_t ws_size,
                              hipStream_t stream) {
    const float* x_in  = (const float*)d_in[0];
    const int*   ei    = (const int*)d_in[1];
    const float* Wconv = (const float*)d_in[2];
    const float* bconv = (const float*)d_in[3];
    const float* Wlin  = (const float*)d_in[4];
    const float* blin  = (const float*)d_in[5];
    const float* ln_g  = (const float*)d_in[6];
    const float* ln_b  = (const float*)d_in[7];
    const float* Wpred = (const float*)d_in[8];
    const float* bpred = (const float*)d_in[9];
    float* out = (float*)d_out;

    const int* src = ei;
    const int* dst = ei + Ee;

    float* ws = (float*)d_ws;
    float* deg   = ws;                         // N
    float* dinv  = deg + Nn;                   // N
    float* h_c   = dinv + Nn;                  // N*H
    float* h_l   = h_c + (size_t)Nn * Hh;      // N*H
    float* agg   = h_l + (size_t)Nn * Hh;      // N*H
    float* x_cur = agg + (size_t)Nn * Hh;      // N*H
    float* x_fin = x_cur + (size_t)Nn * Hh;    // N*H
    float* Wb    = x_fin + (size_t)Nn * Hh;    // 6 * 128*128

    // one-time per call: degrees, dinv, weight swizzle
    k_deg_init<<<(Nn + 255) / 256, 256, 0, stream>>>(deg);
    k_deg_count<<<(Ee + 255) / 256, 256, 0, stream>>>(dst, deg);
    k_dinv<<<(Nn + 255) / 256, 256, 0, stream>>>(deg, dinv);
    k_wswizzle<<<(6 * 8192 + 255) / 256, 256, 0, stream>>>(Wconv, Wlin, Wb);

    const float* xc = x_in;
    for (int l = 0; l < Ll; ++l) {
        k_gemm_dual<<<Nn / 16, 256, 0, stream>>>(
            xc, Wb + (size_t)(l * 2 + 0) * 16384, Wb + (size_t)(l * 2 + 1) * 16384,
            blin + l * Hh, bconv + l * Hh, dinv, h_c, h_l, agg);
        k_edge_scatter<<<(Ee * 32 + 255) / 256, 256, 0, stream>>>(
            src, dst, dinv, h_c, agg);
        k_post_ln<<<(Nn + 7) / 8, 256, 0, stream>>>(
            agg, h_l, ln_g + l * Hh, ln_b + l * Hh, x_cur, x_fin, l == 0);
        xc = x_cur;
    }
    k_pred<<<Nn, 64, 0, stream>>>(x_fin, Wpred, bpred, out);
}